// CAREGNNLayer_78632261255938
// MI455X (gfx1250) — compile-verified
//
#include <hip/hip_runtime.h>
#include <hip/hip_bf16.h>

typedef __attribute__((ext_vector_type(2))) float v2f;
typedef __attribute__((ext_vector_type(8))) float v8f;

#define N_NODES 50000
#define D 128
#define N_REL 3
#define N_EDGES 800000

// workspace layout (float offsets)
#define OFF_PREAGG   0u            // 3*50000*128 = 19,200,000
#define OFF_COMBINED 19200000u     // 6,400,000
#define OFF_GATE     25600000u     // 6,400,000
#define OFF_WSUM     32000000u     // 150,000
#define OFF_RELW     32150000u     // 150,000
#define OFF_ATT      32300000u     // 50,000   (total 32,350,000 floats = 129.4 MB)

__global__ __launch_bounds__(256) void k_zero4(float4* p, int n4) {
  int i = blockIdx.x * blockDim.x + threadIdx.x;
  if (i < n4) p[i] = make_float4(0.f, 0.f, 0.f, 0.f);
}

// Per-node small math: class_probs (softmax-2), relation_weights (softmax-3),
// node attention (relu MLP + dot). One wave per node, 8 nodes per block.
__global__ __launch_bounds__(256) void k_node_small(
    const float* __restrict__ feat,
    const float* __restrict__ pred_w, const float* __restrict__ pred_b,
    const float* __restrict__ att_w1, const float* __restrict__ att_b1,
    const float* __restrict__ att_w2, const float* __restrict__ att_b2,
    const float* __restrict__ relw_w, const float* __restrict__ relw_b,
    float* __restrict__ class_probs, float* __restrict__ node_att,
    float* __restrict__ rel_weights)
{
  __shared__ float fs[8][D];
  const int wid  = threadIdx.x >> 5;
  const int lane = threadIdx.x & 31;
  const int node = blockIdx.x * 8 + wid;   // grid = 6250 -> exactly 50000 nodes

  const float4 f4 = *reinterpret_cast<const float4*>(feat + (size_t)node * D + lane * 4);
  fs[wid][lane*4+0] = f4.x; fs[wid][lane*4+1] = f4.y;
  fs[wid][lane*4+2] = f4.z; fs[wid][lane*4+3] = f4.w;
  __syncthreads();

  // prediction logits + relation-weight logits (per-lane partials over 4 dims)
  float l0 = 0.f, l1 = 0.f, r0 = 0.f, r1 = 0.f, r2 = 0.f;
#pragma unroll
  for (int j = 0; j < 4; ++j) {
    const int d = lane * 4 + j;
    const float fv = fs[wid][d];
    l0 = fmaf(fv, pred_w[d*2+0], l0);
    l1 = fmaf(fv, pred_w[d*2+1], l1);
    r0 = fmaf(fv, relw_w[d*3+0], r0);
    r1 = fmaf(fv, relw_w[d*3+1], r1);
    r2 = fmaf(fv, relw_w[d*3+2], r2);
  }
#pragma unroll
  for (int off = 16; off > 0; off >>= 1) {
    l0 += __shfl_xor(l0, off); l1 += __shfl_xor(l1, off);
    r0 += __shfl_xor(r0, off); r1 += __shfl_xor(r1, off); r2 += __shfl_xor(r2, off);
  }
  l0 += pred_b[0]; l1 += pred_b[1];
  const float mx = fmaxf(l0, l1);
  const float e0 = expf(l0 - mx), e1 = expf(l1 - mx);
  const float pin = 1.f / (e0 + e1);
  const float c0 = e0 * pin, c1 = e1 * pin;

  r0 += relw_b[0]; r1 += relw_b[1]; r2 += relw_b[2];
  const float rmx = fmaxf(fmaxf(r0, r1), r2);
  float g0 = expf(r0 - rmx), g1 = expf(r1 - rmx), g2 = expf(r2 - rmx);
  const float gin = 1.f / (g0 + g1 + g2);
  g0 *= gin; g1 *= gin; g2 *= gin;

  // attention: 128 hidden units total (2 classes x 64); 4 units per lane
  float s0 = 0.f, s1 = 0.f;
#pragma unroll
  for (int t = 0; t < 4; ++t) {
    const int u = lane + 32 * t;
    const int c = u >> 6, h = u & 63;
    const float* w1 = att_w1 + c * (D * 64) + h;   // stride 64 over d
    float acc = att_b1[c * 64 + h];
    for (int d = 0; d < D; ++d) acc = fmaf(fs[wid][d], w1[d * 64], acc);
    acc = fmaxf(acc, 0.f);
    const float ctr = acc * att_w2[c * 64 + h];
    if (c == 0) s0 += ctr; else s1 += ctr;
  }
#pragma unroll
  for (int off = 16; off > 0; off >>= 1) {
    s0 += __shfl_xor(s0, off); s1 += __shfl_xor(s1, off);
  }
  const float natt = (s0 + att_b2[0]) * c0 + (s1 + att_b2[1]) * c1;

  if (lane == 0) {
    class_probs[node*2+0] = c0; class_probs[node*2+1] = c1;
    node_att[node] = natt;
    rel_weights[node*3+0] = g0; rel_weights[node*3+1] = g1; rel_weights[node*3+2] = g2;
  }
}

// Scatter-add: pre_agg[r][dst] += w_e * features[src]; wsum[r][dst] += w_e.
// One wave per edge (4 floats per lane), 8 edges per 256-thread block.
__global__ __launch_bounds__(256) void k_scatter(
    const float* __restrict__ feat, const int* __restrict__ eidx,
    const float* __restrict__ ew, float* __restrict__ pre_agg,
    float* __restrict__ wsum)
{
  const int wid  = threadIdx.x >> 5;
  const int lane = threadIdx.x & 31;
  const long long e = (long long)blockIdx.x * 8 + wid;      // 0 .. 2,399,999
  const int r  = (int)(e / N_EDGES);
  const int el = (int)(e - (long long)r * N_EDGES);
  const int src = eidx[(size_t)r * 2 * N_EDGES + el];
  const int dst = eidx[(size_t)r * 2 * N_EDGES + N_EDGES + el];
  const float w = ew[(size_t)r * N_EDGES + el];

  const float4 f = *reinterpret_cast<const float4*>(feat + (size_t)src * D + lane * 4);
  float* base = pre_agg + ((size_t)r * N_NODES + dst) * D + lane * 4;
  atomicAdd(base + 0, w * f.x);
  atomicAdd(base + 1, w * f.y);
  atomicAdd(base + 2, w * f.z);
  atomicAdd(base + 3, w * f.w);
  if (lane == 0) atomicAdd(&wsum[(size_t)r * N_NODES + dst], w);
}

// Scale pre_agg[r][n] by relation_weights[n][r] (in place) and initialize
// combined[n] = sum_r rw[n][r] * wsum[r][n] * rel_b[r].
__global__ __launch_bounds__(256) void k_scale_bias(
    float* __restrict__ pre_agg, const float* __restrict__ wsum,
    const float* __restrict__ relw, const float* __restrict__ rel_b,
    float* __restrict__ combined)
{
  const int t = blockIdx.x * blockDim.x + threadIdx.x;  // N_NODES * 32
  const int n = t >> 5;
  const int d0 = (t & 31) * 4;
  float4 acc = make_float4(0.f, 0.f, 0.f, 0.f);
#pragma unroll
  for (int r = 0; r < N_REL; ++r) {
    const float rw = relw[n * 3 + r];
    const float w  = wsum[(size_t)r * N_NODES + n];
    float4* p = reinterpret_cast<float4*>(pre_agg + ((size_t)r * N_NODES + n) * D + d0);
    float4 v = *p;
    v.x *= rw; v.y *= rw; v.z *= rw; v.w *= rw;
    *p = v;
    const float4 b = *reinterpret_cast<const float4*>(rel_b + r * D + d0);
    const float s = rw * w;
    acc.x = fmaf(s, b.x, acc.x); acc.y = fmaf(s, b.y, acc.y);
    acc.z = fmaf(s, b.z, acc.z); acc.w = fmaf(s, b.w, acc.w);
  }
  *reinterpret_cast<float4*>(combined + (size_t)n * D + d0) = acc;
}

// WMMA f32 GEMM: C[M x 128] (+)= A[M x 128] @ B[128 x 128] (+ bias).
// 256 threads = 8 waves; block owns a 16x128 C tile; wave w owns cols 16w..16w+15.
// A 16x4 f32 layout: lane<16 -> M=lane, K={k,k+1}; lane>=16 -> M=lane-16, K={k+2,k+3}.
__global__ __launch_bounds__(256) void k_gemm_wmma(
    const float* __restrict__ A, const float* __restrict__ B,
    float* __restrict__ C, const float* __restrict__ bias, int accumulate)
{
  __shared__ float As[16 * D];
  const int tile = blockIdx.x;               // 3125 tiles of 16 rows
  for (int i = threadIdx.x; i < 16 * D; i += 256)
    As[i] = A[(size_t)tile * 16 * D + i];
  __syncthreads();

  const int lane = threadIdx.x & 31;
  const int wave = threadIdx.x >> 5;
  const int hi   = lane >> 4;                // lane-half selects K pair
  const int m    = lane & 15;                // M for A reads / N for B,C
  const int col  = wave * 16 + m;

  v8f c = {};
  for (int k = 0; k < D; k += 4) {
    const int kk = k + 2 * hi;
    const float2 av = *reinterpret_cast<const float2*>(&As[m * D + kk]);
    v2f a; a.x = av.x; a.y = av.y;
    v2f b; b.x = B[kk * D + col]; b.y = B[(kk + 1) * D + col];
    c = __builtin_amdgcn_wmma_f32_16x16x4_f32(false, a, false, b,
                                              (short)0, c, false, false);
  }

#pragma unroll
  for (int v = 0; v < 8; ++v) {
    const int row = tile * 16 + v + 8 * hi;
    float val = c[v];
    if (bias) val += bias[col];
    const size_t idx = (size_t)row * D + col;
    if (accumulate) val += C[idx];
    C[idx] = val;
  }
}

// gated = sigmoid(gate_lin) * combined;  weighted_rel = gated * node_att[n]
// (written back over gate_lin buffer)
__global__ __launch_bounds__(256) void k_gate(
    float* __restrict__ gate_lin, const float* __restrict__ combined,
    const float* __restrict__ natt)
{
  const int t = blockIdx.x * blockDim.x + threadIdx.x;  // N_NODES * 32
  const int n = t >> 5;
  const int d0 = (t & 31) * 4;
  const float a = natt[n];
  float4 g  = *reinterpret_cast<float4*>(gate_lin + (size_t)n * D + d0);
  const float4 cm = *reinterpret_cast<const float4*>(combined + (size_t)n * D + d0);
  float4 o;
  o.x = (1.f / (1.f + expf(-g.x))) * cm.x * a;
  o.y = (1.f / (1.f + expf(-g.y))) * cm.y * a;
  o.z = (1.f / (1.f + expf(-g.z))) * cm.z * a;
  o.w = (1.f / (1.f + expf(-g.w))) * cm.w * a;
  *reinterpret_cast<float4*>(gate_lin + (size_t)n * D + d0) = o;
}

// out = LayerNorm(relu(fused_lin) + transformed) * ln_g + ln_b.  One wave/node.
__global__ __launch_bounds__(256) void k_final(
    const float* __restrict__ fused_lin, const float* __restrict__ transformed,
    const float* __restrict__ ln_g, const float* __restrict__ ln_b,
    float* __restrict__ out)
{
  const int wid  = threadIdx.x >> 5;
  const int lane = threadIdx.x & 31;
  const int node = blockIdx.x * 8 + wid;
  const int d0 = lane * 4;
  const float4 f  = *reinterpret_cast<const float4*>(fused_lin + (size_t)node * D + d0);
  const float4 tr = *reinterpret_cast<const float4*>(transformed + (size_t)node * D + d0);
  float v[4];
  v[0] = fmaxf(f.x, 0.f) + tr.x;
  v[1] = fmaxf(f.y, 0.f) + tr.y;
  v[2] = fmaxf(f.z, 0.f) + tr.z;
  v[3] = fmaxf(f.w, 0.f) + tr.w;
  float s  = v[0] + v[1] + v[2] + v[3];
  float s2 = v[0]*v[0] + v[1]*v[1] + v[2]*v[2] + v[3]*v[3];
#pragma unroll
  for (int off = 16; off > 0; off >>= 1) {
    s  += __shfl_xor(s,  off);
    s2 += __shfl_xor(s2, off);
  }
  const float mu  = s * (1.f / 128.f);
  const float var = s2 * (1.f / 128.f) - mu * mu;
  const float rs  = rsqrtf(var + 1e-5f);
#pragma unroll
  for (int j = 0; j < 4; ++j) {
    const int d = d0 + j;
    out[(size_t)node * D + d] = (v[j] - mu) * rs * ln_g[d] + ln_b[d];
  }
}

extern "C" void kernel_launch(void* const* d_in, const int* in_sizes, int n_in,
                              void* d_out, int out_size, void* d_ws, size_t ws_size,
                              hipStream_t stream) {
  (void)in_sizes; (void)n_in; (void)out_size; (void)ws_size;
  const float* feat   = (const float*)d_in[0];
  const int*   eidx   = (const int*)  d_in[1];
  const float* ew     = (const float*)d_in[2];
  const float* att_w1 = (const float*)d_in[3];
  const float* att_b1 = (const float*)d_in[4];
  const float* att_w2 = (const float*)d_in[5];
  const float* att_b2 = (const float*)d_in[6];
  const float* pred_w = (const float*)d_in[7];
  const float* pred_b = (const float*)d_in[8];
  const float* rel_w  = (const float*)d_in[9];
  const float* rel_b  = (const float*)d_in[10];
  const float* relw_w = (const float*)d_in[11];
  const float* relw_b = (const float*)d_in[12];
  const float* gate_w = (const float*)d_in[13];
  const float* gate_b = (const float*)d_in[14];
  const float* self_w = (const float*)d_in[15];
  const float* self_b = (const float*)d_in[16];
  const float* feat_w = (const float*)d_in[17];
  const float* feat_b = (const float*)d_in[18];
  const float* fus_w  = (const float*)d_in[19];
  const float* fus_b  = (const float*)d_in[20];
  const float* ln_g   = (const float*)d_in[21];
  const float* ln_b   = (const float*)d_in[22];

  float* ws       = (float*)d_ws;
  float* pre_agg  = ws + OFF_PREAGG;
  float* combined = ws + OFF_COMBINED;
  float* gatebuf  = ws + OFF_GATE;
  float* wsum     = ws + OFF_WSUM;
  float* relw     = ws + OFF_RELW;
  float* natt     = ws + OFF_ATT;

  float* out_main  = (float*)d_out;                       // 50000 x 128
  float* out_probs = out_main + (size_t)N_NODES * D;      // 50000 x 2

  // 1) zero accumulators
  k_zero4<<<(4800000 + 255) / 256, 256, 0, stream>>>((float4*)pre_agg, 4800000);
  k_zero4<<<(37500 + 255) / 256, 256, 0, stream>>>((float4*)wsum, 37500);

  // 2) per-node small math (class_probs -> output tail, node_att, rel weights)
  k_node_small<<<N_NODES / 8, 256, 0, stream>>>(feat, pred_w, pred_b, att_w1, att_b1,
                                                att_w2, att_b2, relw_w, relw_b,
                                                out_probs, natt, relw);

  // 3) edge scatter (weighted feature aggregation, L2-resident)
  k_scatter<<<(N_REL * N_EDGES) / 8, 256, 0, stream>>>(feat, eidx, ew, pre_agg, wsum);

  // 4) fold relation weights + rel_b bias term; init combined
  k_scale_bias<<<(N_NODES * 32) / 256, 256, 0, stream>>>(pre_agg, wsum, relw, rel_b, combined);

  // 5) combined += scaled_pre_agg[r] @ rel_w[r]   (3 WMMA GEMMs)
  for (int r = 0; r < N_REL; ++r)
    k_gemm_wmma<<<N_NODES / 16, 256, 0, stream>>>(pre_agg + (size_t)r * N_NODES * D,
                                                  rel_w + (size_t)r * D * D,
                                                  combined, nullptr, 1);

  // 6) gate_lin = combined @ gate_w + gate_b; then gated * node_att -> gatebuf
  k_gemm_wmma<<<N_NODES / 16, 256, 0, stream>>>(combined, gate_w, gatebuf, gate_b, 0);
  k_gate<<<(N_NODES * 32) / 256, 256, 0, stream>>>(gatebuf, combined, natt);

  // 7) self / feat transforms (reuse pre_agg space) and fused MLP
  float* self_out    = pre_agg;
  float* transformed = pre_agg + (size_t)N_NODES * D;
  float* fusedbuf    = pre_agg + (size_t)2 * N_NODES * D;
  k_gemm_wmma<<<N_NODES / 16, 256, 0, stream>>>(feat, self_w, self_out, self_b, 0);
  k_gemm_wmma<<<N_NODES / 16, 256, 0, stream>>>(feat, feat_w, transformed, feat_b, 0);
  k_gemm_wmma<<<N_NODES / 16, 256, 0, stream>>>(self_out, fus_w, fusedbuf, fus_b, 0);
  k_gemm_wmma<<<N_NODES / 16, 256, 0, stream>>>(gatebuf, fus_w + D * D, fusedbuf, nullptr, 1);

  // 8) relu + residual + layernorm -> main output
  k_final<<<N_NODES / 8, 256, 0, stream>>>(fusedbuf, transformed, ln_g, ln_b, out_main);
}